// DTSP_GNN_Prates_86612310491809
// MI455X (gfx1250) — compile-verified
//
#include <hip/hip_runtime.h>

typedef __attribute__((ext_vector_type(2))) float v2f;
typedef __attribute__((ext_vector_type(8))) float v8f;

// out[i,:] = node[i,:] @ Wn + bn    ([N,2] @ [2,2] + [2])
__global__ void node_linear_kernel(const float* __restrict__ nf,
                                   const float* __restrict__ Wn,
                                   const float* __restrict__ bn,
                                   float* __restrict__ out, int N) {
  int i = blockIdx.x * blockDim.x + threadIdx.x;
  if (i >= N) return;
  float x0 = nf[2 * i + 0], x1 = nf[2 * i + 1];
  float o0 = fmaf(x1, Wn[2], fmaf(x0, Wn[0], bn[0]));
  float o1 = fmaf(x1, Wn[3], fmaf(x0, Wn[1], bn[1]));
  out[2 * i + 0] = o0;
  out[2 * i + 1] = o1;
}

// Fused edge MLP over FULL 16-edge tiles:
//   ef = e @ (We @ Wu[0:10]) + s @ Wu[10:12] + d @ Wu[12:14] + (be @ Wu[0:10] + bu)
// Two chained V_WMMA_F32_16X16X4_F32 per tile (K = [d0,d1,0,0], then [e0,e1,s0,s1]).
__global__ void edge_mlp_wmma_kernel(const float* __restrict__ nf,
                                     const float* __restrict__ ef_in,
                                     const int* __restrict__ eidx,
                                     const float* __restrict__ We,
                                     const float* __restrict__ be,
                                     const float* __restrict__ Wu,
                                     const float* __restrict__ bu,
                                     float* __restrict__ ef_out,
                                     int E, int nfull) {
  const int lane = threadIdx.x & 31;
  const int n    = lane & 15;   // output column (and M-slot for A assembly)
  const int half = lane >> 4;   // 0: lanes 0-15, 1: lanes 16-31

  // ---- Fused weights for this lane's column n (hoisted, runs once) ----
  float wfe0 = 0.f, wfe1 = 0.f, bias = 0.f;
  float ws0 = 0.f, ws1 = 0.f, wd0 = 0.f, wd1 = 0.f;
  if (n < 10) {
    bias = bu[n];
#pragma unroll
    for (int j = 0; j < 10; ++j) {
      float wj = Wu[j * 10 + n];         // Wu is (14,10) row-major
      wfe0 = fmaf(We[j],      wj, wfe0); // We row 0 (We is (2,10) row-major)
      wfe1 = fmaf(We[10 + j], wj, wfe1); // We row 1
      bias = fmaf(be[j],      wj, bias);
    }
    ws0 = Wu[10 * 10 + n];
    ws1 = Wu[11 * 10 + n];
    wd0 = Wu[12 * 10 + n];
    wd1 = Wu[13 * 10 + n];
  }

  // B operands (4x16 f32): VGPR0 holds rows K0|K2 across lane halves, VGPR1 rows K1|K3.
  v2f B1, B2;
  B1.x = half ? ws0 : wfe0;  // K0 = (We@Wu0) row0 ; K2 = Ws row0
  B1.y = half ? ws1 : wfe1;  // K1 = (We@Wu0) row1 ; K3 = Ws row1
  B2.x = half ? 0.f : wd0;   // K4 = Wd row0       ; K6 = 0 (pad)
  B2.y = half ? 0.f : wd1;   // K5 = Wd row1       ; K7 = 0 (pad)

  const int* rowp = eidx;      // edge_index[0, :]
  const int* colp = eidx + E;  // edge_index[1, :]

  const int waves_per_blk = blockDim.x >> 5;
  const int wave   = blockIdx.x * waves_per_blk + (threadIdx.x >> 5);
  const int nwaves = gridDim.x * waves_per_blk;

  for (int t = wave; t < nfull; t += nwaves) {
    const int eid = (t << 4) + n;   // full tile: always in range

    // A operand (16x4 f32): lane<16 = M,{K0,K1}; lane>=16 = M,{K2,K3}.
    // Lanes 0-15 fetch edge features + dst gather; lanes 16-31 fetch src gather.
    v2f A1, A2;
    if (half == 0) {
      const v2f e = *(const v2f*)(ef_in + 2 * (size_t)eid);
      const int c = colp[eid];
      const v2f d = *(const v2f*)(nf + 2 * (size_t)c);
      A1 = e;                    // K0 = e0, K1 = e1
      A2 = d;                    // K4 = d0, K5 = d1
    } else {
      const int r = rowp[eid];
      const v2f s = *(const v2f*)(nf + 2 * (size_t)r);
      A1 = s;                    // K2 = s0, K3 = s1
      A2.x = 0.f; A2.y = 0.f;    // K6 = K7 = 0
    }

    v8f acc;
#pragma unroll
    for (int r = 0; r < 8; ++r) acc[r] = bias;

    // D = A1*B1 + (A2*B2 + bias)
    acc = __builtin_amdgcn_wmma_f32_16x16x4_f32(false, A2, false, B2,
                                                (short)0, acc, false, false);
    acc = __builtin_amdgcn_wmma_f32_16x16x4_f32(false, A1, false, B1,
                                                (short)0, acc, false, false);

    // C/D layout: VGPR r -> row (half*8 + r), column n.
    // Single base pointer; 8 stores with immediate offsets r*40 bytes.
    if (n < 10) {
      float* p = ef_out + ((size_t)(t << 4) + (half << 3)) * 10 + n;
#pragma unroll
      for (int r = 0; r < 8; ++r) p[r * 10] = acc[r];
    }
  }
}

// Scalar tail: edges [base, E) computed directly (two-stage affine, f32 VALU).
__global__ void edge_tail_kernel(const float* __restrict__ nf,
                                 const float* __restrict__ ef_in,
                                 const int* __restrict__ eidx,
                                 const float* __restrict__ We,
                                 const float* __restrict__ be,
                                 const float* __restrict__ Wu,
                                 const float* __restrict__ bu,
                                 float* __restrict__ ef_out,
                                 int E, int base) {
  int m = base + blockIdx.x * blockDim.x + threadIdx.x;
  if (m >= E) return;
  float e0 = ef_in[2 * m + 0], e1 = ef_in[2 * m + 1];
  int r = eidx[m], c = eidx[E + m];
  float s0 = nf[2 * (size_t)r + 0], s1 = nf[2 * (size_t)r + 1];
  float d0 = nf[2 * (size_t)c + 0], d1 = nf[2 * (size_t)c + 1];
#pragma unroll
  for (int j = 0; j < 10; ++j) {
    float h = fmaf(e1, We[10 + j], fmaf(e0, We[j], be[j]));  // stage-1 feature j
    // accumulate stage-2 partial for every output column
    if (j == 0) {
#pragma unroll
      for (int k = 0; k < 10; ++k)
        ef_out[(size_t)m * 10 + k] = fmaf(h, Wu[0 * 10 + k], bu[k]);
    } else {
#pragma unroll
      for (int k = 0; k < 10; ++k)
        ef_out[(size_t)m * 10 + k] = fmaf(h, Wu[j * 10 + k], ef_out[(size_t)m * 10 + k]);
    }
  }
#pragma unroll
  for (int k = 0; k < 10; ++k) {
    float v = ef_out[(size_t)m * 10 + k];
    v = fmaf(s0, Wu[10 * 10 + k], v);
    v = fmaf(s1, Wu[11 * 10 + k], v);
    v = fmaf(d0, Wu[12 * 10 + k], v);
    v = fmaf(d1, Wu[13 * 10 + k], v);
    ef_out[(size_t)m * 10 + k] = v;
  }
}

extern "C" void kernel_launch(void* const* d_in, const int* in_sizes, int n_in,
                              void* d_out, int out_size, void* d_ws, size_t ws_size,
                              hipStream_t stream) {
  const float* nf  = (const float*)d_in[0];  // [N,2]
  const float* efi = (const float*)d_in[1];  // [E,2]
  const int*   eix = (const int*)  d_in[2];  // [2,E] int32
  const float* We  = (const float*)d_in[3];  // [2,10]
  const float* be  = (const float*)d_in[4];  // [10]
  const float* Wu  = (const float*)d_in[5];  // [14,10]
  const float* bu  = (const float*)d_in[6];  // [10]
  const float* Wn  = (const float*)d_in[7];  // [2,2]
  const float* bn  = (const float*)d_in[8];  // [2]

  const int N = in_sizes[0] / 2;
  const int E = in_sizes[1] / 2;

  float* out_nodes = (float*)d_out;                 // [N,2] first in tuple order
  float* out_edges = out_nodes + (size_t)2 * N;     // [E,10]

  node_linear_kernel<<<(N + 255) / 256, 256, 0, stream>>>(nf, Wn, bn, out_nodes, N);

  const int nfull = E >> 4;   // full 16-edge tiles (branch-free WMMA path)
  if (nfull > 0) {
    // 1024 blocks x 8 waves = 8192 waves over 200K tiles (~24 tiles/wave):
    // enough in-flight gathers to cover L2 latency and saturate the 23.3 TB/s
    // HBM write stream (128 MB of ef output dominates).
    edge_mlp_wmma_kernel<<<1024, 256, 0, stream>>>(nf, efi, eix, We, be, Wu, bu,
                                                   out_edges, E, nfull);
  }
  const int tail = E - (nfull << 4);
  if (tail > 0) {
    edge_tail_kernel<<<1, 32, 0, stream>>>(nf, efi, eix, We, be, Wu, bu,
                                           out_edges, E, nfull << 4);
  }
}